// MeanAggregator_13846974562846
// MI455X (gfx1250) — compile-verified
//
#include <hip/hip_runtime.h>
#include <stdint.h>

// ---------------------------------------------------------------------------
// MeanAggregator: out[b,:] = mean_{s<10} table[idx[b,s], :]   (f32, d=128)
// Memory-bound gather (~570 MB @ 23.3 TB/s => ~24us floor). Uses CDNA5 TDM
// gather-mode (tensor_load_to_lds) to DMA indexed 512B rows into LDS.
// ---------------------------------------------------------------------------

typedef unsigned int uint32;
typedef uint32 u32x4 __attribute__((ext_vector_type(4)));
typedef int    i32x8 __attribute__((ext_vector_type(8)));
typedef int    i32x4 __attribute__((ext_vector_type(4)));
typedef float  f32x4 __attribute__((ext_vector_type(4)));

#define FEAT_DIM        128
#define NSAMP           10
#define ROW_BYTES       (FEAT_DIM * 4)          // 512 B per table row
#define WAVES_PER_BLOCK 8
#define BLOCK_THREADS   (WAVES_PER_BLOCK * 32)
#define N_NODES_DIM     1000000u

// Issue one TDM gather-mode load: fetch `nvalid` rows (indices in r03/r47)
// of FEAT_DIM f32 each from `base` into LDS at byte offset `lds_off`.
// D# layout per CDNA5 ISA ch.8 (gather_mode=1, 32-bit indices).
__device__ __forceinline__ void tdm_gather8(const float* base, uint32 lds_off,
                                            int nvalid, i32x4 r03, i32x4 r47) {
  const uint64_t ga = (uint64_t)(uintptr_t)base;

  u32x4 g0;
  g0[0] = 1u                  // count = 1 (valid descriptor)
        | (1u << 30)          // gather_index_size = 32-bit
        | (1u << 31);         // gather_mode = 1
  g0[1] = lds_off;            // lds_addr (bytes, wg-relative)
  g0[2] = (uint32)ga;                                   // global_addr[31:0]
  g0[3] = ((uint32)(ga >> 32) & 0x01FFFFFFu)            // global_addr[56:32]
        | (2u << 30);                                   // type = 2 ("image")

  i32x8 g1;
  g1[0] = (2 << 16);                                    // data_size = 4 bytes
  g1[1] = (int)(((uint32)FEAT_DIM & 0xFFFFu) << 16);    // tensor_dim0[15:0]
  g1[2] = (int)((N_NODES_DIM & 0xFFFFu) << 16);         // tensor_dim1[15:0]
  g1[3] = (int)((N_NODES_DIM >> 16) & 0xFFFFu)          // tensor_dim1[31:16]
        | (FEAT_DIM << 16);                             // tile_dim0 = 128
  g1[4] = nvalid & 0xFFFF;                              // tile_dim1 = #indices
  g1[5] = FEAT_DIM;                                     // tensor_dim0_stride lo32
  g1[6] = 0;                                            // stride hi / dim1_stride (ignored)
  g1[7] = 0;

  i32x8 gx;                                             // extra operand slot on
  gx[0] = 0; gx[1] = 0; gx[2] = 0; gx[3] = 0;           // clang-23's 6-arg form;
  gx[4] = 0; gx[5] = 0; gx[6] = 0; gx[7] = 0;           // zero-filled

  __builtin_amdgcn_tensor_load_to_lds(g0, g1, r03, r47, gx, 0);
}

__global__ __launch_bounds__(BLOCK_THREADS)
void MeanAggregator_13846974562846_kernel(const float* __restrict__ table,
                                          const int* __restrict__ nidx,
                                          float* __restrict__ out,
                                          int batch) {
  __shared__ char smem[WAVES_PER_BLOCK * NSAMP * ROW_BYTES];  // 40 KB

  const int lane = threadIdx.x & 31;
  const int wave = threadIdx.x >> 5;
  const int b    = blockIdx.x * WAVES_PER_BLOCK + wave;       // uniform per wave
  if (b >= batch) return;

  // Lanes 0..9 load this row's 10 neighbor indices (one coalesced request),
  // then broadcast each into scalar registers for descriptor construction.
  int my = 0;
  if (lane < NSAMP) my = nidx[b * NSAMP + lane];
  const int i0 = __builtin_amdgcn_readlane(my, 0);
  const int i1 = __builtin_amdgcn_readlane(my, 1);
  const int i2 = __builtin_amdgcn_readlane(my, 2);
  const int i3 = __builtin_amdgcn_readlane(my, 3);
  const int i4 = __builtin_amdgcn_readlane(my, 4);
  const int i5 = __builtin_amdgcn_readlane(my, 5);
  const int i6 = __builtin_amdgcn_readlane(my, 6);
  const int i7 = __builtin_amdgcn_readlane(my, 7);
  const int i8 = __builtin_amdgcn_readlane(my, 8);
  const int i9 = __builtin_amdgcn_readlane(my, 9);

  char* stage = &smem[wave * (NSAMP * ROW_BYTES)];
  const uint32 lds0 = (uint32)(uintptr_t)stage;

  // Two in-flight gather DMAs per wave: rows 0..7, then rows 8..9.
  i32x4 rA; rA[0] = i0; rA[1] = i1; rA[2] = i2; rA[3] = i3;
  i32x4 rB; rB[0] = i4; rB[1] = i5; rB[2] = i6; rB[3] = i7;
  tdm_gather8(table, lds0, 8, rA, rB);
  i32x4 rC; rC[0] = i8; rC[1] = i9; rC[2] = 0; rC[3] = 0;
  i32x4 rZ; rZ[0] = 0;  rZ[1] = 0;  rZ[2] = 0; rZ[3] = 0;
  tdm_gather8(table, lds0 + 8u * ROW_BYTES, 2, rC, rZ);

  __builtin_amdgcn_s_wait_tensorcnt(0);

  // Each lane owns 4 contiguous feature floats (16B): sum 10 staged rows.
  const f32x4* rows = (const f32x4*)stage;     // row stride = FEAT_DIM/4 vecs
  f32x4 acc = rows[lane];
#pragma unroll
  for (int s = 1; s < NSAMP; ++s)
    acc += rows[s * (FEAT_DIM / 4) + lane];
  acc *= 0.1f;

  // Non-temporal 128b store: keep the 51MB output stream out of L2 so the
  // gather-serving table lines stay resident.
  f32x4* op = (f32x4*)(out + (size_t)b * FEAT_DIM);
  __builtin_nontemporal_store(acc, op + lane);
}

extern "C" void kernel_launch(void* const* d_in, const int* in_sizes, int n_in,
                              void* d_out, int out_size, void* d_ws, size_t ws_size,
                              hipStream_t stream) {
  const float* table = (const float*)d_in[0];   // [N_NODES, 128] f32
  const int*   nidx  = (const int*)d_in[1];     // [BATCH, 10] i32
  float*       out   = (float*)d_out;           // [BATCH, 128] f32

  const int batch  = in_sizes[1] / NSAMP;
  const int blocks = (batch + WAVES_PER_BLOCK - 1) / WAVES_PER_BLOCK;

  MeanAggregator_13846974562846_kernel<<<blocks, BLOCK_THREADS, 0, stream>>>(
      table, nidx, out, batch);
}